// TripletLoss_85598698209924
// MI455X (gfx1250) — compile-verified
//
#include <hip/hip_runtime.h>

typedef __attribute__((ext_vector_type(2))) float v2f;
typedef __attribute__((ext_vector_type(8))) float v8f;

#define BATCH 384
#define KDIM  1024
#define MARGIN_V 1.0f

// ---------------------------------------------------------------------------
// K1: sq[row] = sum_k E[row,k]^2   (one block per row, deterministic reduce)
// ---------------------------------------------------------------------------
__global__ void sqnorm_kernel(const float* __restrict__ E, float* __restrict__ sq) {
    __shared__ float red[256];
    const int row = blockIdx.x;
    const int tid = threadIdx.x;
    const float* p = E + (size_t)row * KDIM;
    float s = 0.0f;
    for (int k = tid; k < KDIM; k += 256) {
        float v = p[k];
        s += v * v;
    }
    red[tid] = s;
    __syncthreads();
    for (int off = 128; off > 0; off >>= 1) {
        if (tid < off) red[tid] += red[tid + off];
        __syncthreads();
    }
    if (tid == 0) sq[row] = red[0];
}

// ---------------------------------------------------------------------------
// K2: D = sq_i + sq_j - 2 * (E E^T), one wave per 16x16 tile via
//     V_WMMA_F32_16X16X4_F32.
// A-layout (16x4 f32): lane L, vgpr v -> M = L&15, K = v + 2*(L>>4)
// B-layout mirrored:    lane L, vgpr v -> N = L&15, K = v + 2*(L>>4)
// C/D layout:           vgpr r        -> M = r + 8*(L>>4), N = L&15
// ---------------------------------------------------------------------------
__global__ void gram_wmma_kernel(const float* __restrict__ E,
                                 const float* __restrict__ sq,
                                 float* __restrict__ D) {
    const int tm   = blockIdx.x;          // tile row (0..23)
    const int tn   = blockIdx.y;          // tile col (0..23)
    const int lane = threadIdx.x;         // 0..31 (one wave per block)
    const int half = lane >> 4;           // 0 or 1
    const int l16  = lane & 15;

    const int rowA = tm * 16 + l16;
    const int rowB = tn * 16 + l16;
    const float* pa = E + (size_t)rowA * KDIM + 2 * half;
    const float* pb = E + (size_t)rowB * KDIM + 2 * half;

    v8f acc = {};
    #pragma unroll 8
    for (int k = 0; k < KDIM; k += 4) {
        v2f a = *(const v2f*)(pa + k);
        v2f b = *(const v2f*)(pb + k);
        // D = A x B + C   (fp32, 16x16x4)
        acc = __builtin_amdgcn_wmma_f32_16x16x4_f32(
            /*neg_a=*/false, a, /*neg_b=*/false, b,
            /*c_mod=*/(short)0, acc, /*reuse_a=*/false, /*reuse_b=*/false);
    }

    const int n   = tn * 16 + l16;
    const float sn = sq[n];
    #pragma unroll
    for (int r = 0; r < 8; ++r) {
        const int m = tm * 16 + r + 8 * half;
        D[(size_t)m * BATCH + n] = sq[m] + sn - 2.0f * acc[r];
    }
}

// ---------------------------------------------------------------------------
// K3: per-anchor triplet accumulation.
// For anchor a: sum over p>a with lab[p]==lab[a], n with lab[n]!=lab[a] of
// relu(D[a,p]-D[a,n]+margin). Deterministic block tree reduction.
// ---------------------------------------------------------------------------
__global__ void triplet_kernel(const float* __restrict__ D,
                               const long long* __restrict__ labels,
                               float* __restrict__ partial_sum,
                               float* __restrict__ partial_cnt) {
    __shared__ float Drow[BATCH];
    __shared__ int   lab[BATCH];
    __shared__ float red_s[256];
    __shared__ float red_c[256];

    const int a   = blockIdx.x;
    const int tid = threadIdx.x;

    for (int i = tid; i < BATCH; i += 256) {
        Drow[i] = D[(size_t)a * BATCH + i];
        lab[i]  = (int)labels[i];
    }
    __syncthreads();

    const int la = lab[a];
    float s = 0.0f;
    float c = 0.0f;
    for (int p = a + 1; p < BATCH; ++p) {
        if (lab[p] != la) continue;
        const float dap = Drow[p];
        for (int n = tid; n < BATCH; n += 256) {
            if (lab[n] == la) continue;
            const float v = dap - Drow[n] + MARGIN_V;
            s += (v > 0.0f) ? v : 0.0f;
            c += 1.0f;
        }
    }

    red_s[tid] = s;
    red_c[tid] = c;
    __syncthreads();
    for (int off = 128; off > 0; off >>= 1) {
        if (tid < off) {
            red_s[tid] += red_s[tid + off];
            red_c[tid] += red_c[tid + off];
        }
        __syncthreads();
    }
    if (tid == 0) {
        partial_sum[a] = red_s[0];
        partial_cnt[a] = red_c[0];
    }
}

// ---------------------------------------------------------------------------
// K4: ordered final reduction -> mean, count
// ---------------------------------------------------------------------------
__global__ void finalize_kernel(const float* __restrict__ partial_sum,
                                const float* __restrict__ partial_cnt,
                                float* __restrict__ out) {
    if (threadIdx.x == 0 && blockIdx.x == 0) {
        float s = 0.0f, c = 0.0f;
        for (int i = 0; i < BATCH; ++i) {
            s += partial_sum[i];
            c += partial_cnt[i];
        }
        out[0] = (c > 0.0f) ? (s / c) : 0.0f;
        out[1] = c;
    }
}

extern "C" void kernel_launch(void* const* d_in, const int* in_sizes, int n_in,
                              void* d_out, int out_size, void* d_ws, size_t ws_size,
                              hipStream_t stream) {
    const float*     E      = (const float*)d_in[0];       // [384,1024] fp32
    const long long* labels = (const long long*)d_in[1];   // [384] int64
    float* out = (float*)d_out;

    // Workspace carve-up (all 256B aligned).
    char* ws = (char*)d_ws;
    float* sq          = (float*)(ws);                               // 384 f32
    float* D           = (float*)(ws + 2048);                        // 384*384 f32
    float* partial_sum = (float*)(ws + 2048 + BATCH * BATCH * 4);    // 384 f32
    float* partial_cnt = partial_sum + 512;                          // 384 f32

    sqnorm_kernel<<<BATCH, 256, 0, stream>>>(E, sq);
    gram_wmma_kernel<<<dim3(BATCH / 16, BATCH / 16), 32, 0, stream>>>(E, sq, D);
    triplet_kernel<<<BATCH, 256, 0, stream>>>(D, labels, partial_sum, partial_cnt);
    finalize_kernel<<<1, 32, 0, stream>>>(partial_sum, partial_cnt, out);
}